// SpikeTemporalConvNet2D_67740224193390
// MI455X (gfx1250) — compile-verified
//
#include <hip/hip_runtime.h>

typedef __attribute__((ext_vector_type(2))) float v2f;
typedef __attribute__((ext_vector_type(8))) float v8f;

#define NP 134400           // 64 * 21 * 100 spatial positions
#define CH64NP (64 * 134400)

__device__ __forceinline__ int imin(int a, int b) { return a < b ? a : b; }

// 16-step LIF forward: reset = (mem>1), mem = 0.99*mem + y - reset, acc += (mem>1)
__device__ __forceinline__ float lif_mean16(float y) {
    float mem = 0.0f, acc = 0.0f;
#pragma unroll
    for (int t = 0; t < 16; ++t) {
        float reset = (mem > 1.0f) ? 1.0f : 0.0f;
        mem = 0.99f * mem + y - reset;       // THR = 1.0
        acc += (mem > 1.0f) ? 1.0f : 0.0f;
    }
    return acc * 0.0625f;                    // / NUM_STEPS
}

// Causal dilated 1x2 conv (TAPS==2) or 1x1 conv (TAPS==1) as f32 WMMA GEMM.
//   out[o, p] = bias[o] + sum_{k'} W[o, k'] * B[k', p]
//   TAPS==2: k' = 2*i + t, B[2i+0, p] = in[i, p - dil] (0 if w<dil), B[2i+1, p] = in[i, p]
//   TAPS==1: k' = i,       B[i, p]    = in[i, p]
// Input addressed as in + c*cstride + n*nstride + h*100 + w.
// Output layout: out[o * NP + p], p = n*2100 + h*100 + w.
// WMMA f32 16x16x4 layouts (ISA 7.12.2): A: M=lane%16, K=2*(lane/16)+v
//                                        B: K=2*(lane/16)+v, N=lane%16
//                                        D: M=v+8*(lane/16), N=lane%16
// CIN/TAPS are compile-time so the K loop fully unrolls into a WMMA stream.
template <int CIN, int TAPS>
__launch_bounds__(32)
__global__ void conv_wmma_kernel(const float* __restrict__ in,
                                 const float* __restrict__ wgt,
                                 const float* __restrict__ bias,
                                 float* __restrict__ out,
                                 int Cout, int dil,
                                 long cstride, long nstride) {
    constexpr int Kp = CIN * TAPS;

    const int lane = threadIdx.x & 31;
    const int nIdx = lane & 15;   // N within tile; also A-matrix row M
    const int hi   = lane >> 4;   // upper K half selector

    const int p = blockIdx.x * 16 + nIdx;
    const int w = p % 100;
    const int r = p / 100;
    const int h = r % 21;
    const int n = r / 21;
    const long posoff  = (long)n * nstride + (long)h * 100 + w;
    const bool canSh   = (w >= dil);
    const long posoffS = canSh ? (posoff - (long)dil) : posoff; // keep addr valid
    const float shMul  = canSh ? 1.0f : 0.0f;

    const int obase = blockIdx.y * 16;
    const float* Arow = wgt + (size_t)imin(obase + nIdx, Cout - 1) * Kp;

    v8f acc = {};
#pragma unroll
    for (int k0 = 0; k0 < Kp; k0 += 4) {
        const int ka = k0 + 2 * hi;                 // this lane's base k'
        v2f a;
        if constexpr (Kp >= 4) {
            a = *(const v2f*)(Arow + ka);           // in-range, 8B-aligned: one b64 load
        } else {                                    // K padded to 4 (CIN==1 case)
            a.x = (ka     < Kp) ? Arow[ka]     : 0.0f;
            a.y = (ka + 1 < Kp) ? Arow[ka + 1] : 0.0f;
        }

        v2f b;
        if constexpr (TAPS == 2) {
            int c = (k0 >> 1) + hi;                 // provably <= CIN-1 when Kp>=4
            if constexpr (Kp < 4) c = imin(c, CIN - 1);
            const float* base = in + (size_t)c * cstride;
            b.x = base[posoffS] * shMul;            // tap 0: x[w - dil]
            b.y = base[posoff];                     // tap 1: x[w]
        } else {
            b.x = in[(size_t)(ka)     * cstride + posoff];
            b.y = in[(size_t)(ka + 1) * cstride + posoff];
        }
        // (neg_a, A, neg_b, B, c_mod, C, reuse_a, reuse_b)
        acc = __builtin_amdgcn_wmma_f32_16x16x4_f32(false, a, false, b,
                                                    (short)0, acc, false, false);
    }

#pragma unroll
    for (int v = 0; v < 8; ++v) {
        const int o = obase + v + 8 * hi;           // D layout: M = v + 8*hi
        if (o < Cout) out[(size_t)o * NP + p] = acc[v] + bias[o];
    }
}

// Per-channel BN stats over NP elements; folds gamma/beta into scale/shift.
__global__ void bn_stats_kernel(const float* __restrict__ x,
                                const float* __restrict__ g,
                                const float* __restrict__ b,
                                float* __restrict__ scale,
                                float* __restrict__ shift) {
    const int c = blockIdx.x;
    const float* xc = x + (size_t)c * NP;
    float s = 0.0f, sq = 0.0f;
    for (int i = threadIdx.x; i < NP; i += blockDim.x) {
        const float v = xc[i];
        s += v; sq += v * v;
    }
    __shared__ float ss[256], sqq[256];
    ss[threadIdx.x] = s; sqq[threadIdx.x] = sq;
    __syncthreads();
    for (int st = 128; st > 0; st >>= 1) {
        if (threadIdx.x < st) {
            ss[threadIdx.x]  += ss[threadIdx.x + st];
            sqq[threadIdx.x] += sqq[threadIdx.x + st];
        }
        __syncthreads();
    }
    if (threadIdx.x == 0) {
        const float mean = ss[0] * (1.0f / NP);
        const float var  = sqq[0] * (1.0f / NP) - mean * mean;
        const float inv  = rsqrtf(var + 1e-5f);
        const float sc   = g[c] * inv;
        scale[c] = sc;
        shift[c] = b[c] - mean * sc;
    }
}

// In-place BN normalize + 16-step LIF mean. grid = (NP/256, C)
__global__ void bn_lif_kernel(float* __restrict__ x,
                              const float* __restrict__ scale,
                              const float* __restrict__ shift) {
    const int c = blockIdx.y;
    const size_t idx = (size_t)c * NP + (size_t)blockIdx.x * blockDim.x + threadIdx.x;
    x[idx] = lif_mean16(x[idx] * scale[c] + shift[c]);
}

// out = lif_mean(a + b). grid = (NP/256, C)
__global__ void add_lif_kernel(const float* __restrict__ a,
                               const float* __restrict__ b,
                               float* __restrict__ o) {
    const size_t idx = (size_t)blockIdx.y * NP + (size_t)blockIdx.x * blockDim.x + threadIdx.x;
    o[idx] = lif_mean16(a[idx] + b[idx]);
}

// Second output: spks[:, :, -1]  (1344 = 64*21 elements)
__global__ void tail_kernel(float* __restrict__ out) {
    const int i = blockIdx.x * blockDim.x + threadIdx.x;
    if (i < 1344) {
        const int n = i / 21, h = i % 21;
        out[NP + i] = out[(size_t)n * 2100 + h * 100 + 99];
    }
}

extern "C" void kernel_launch(void* const* d_in, const int* in_sizes, int n_in,
                              void* d_out, int out_size, void* d_ws, size_t ws_size,
                              hipStream_t stream) {
    const float* x     = (const float*)d_in[0];
    const float* cw1_0 = (const float*)d_in[1];
    const float* cb1_0 = (const float*)d_in[2];
    const float* g1_0  = (const float*)d_in[3];
    const float* b1_0  = (const float*)d_in[4];
    const float* cw2_0 = (const float*)d_in[5];
    const float* cb2_0 = (const float*)d_in[6];
    const float* g2_0  = (const float*)d_in[7];
    const float* b2_0  = (const float*)d_in[8];
    const float* dsw_0 = (const float*)d_in[9];
    const float* dsb_0 = (const float*)d_in[10];
    const float* cw1_1 = (const float*)d_in[11];
    const float* cb1_1 = (const float*)d_in[12];
    const float* g1_1  = (const float*)d_in[13];
    const float* b1_1  = (const float*)d_in[14];
    const float* cw2_1 = (const float*)d_in[15];
    const float* cb2_1 = (const float*)d_in[16];
    const float* g2_1  = (const float*)d_in[17];
    const float* b2_1  = (const float*)d_in[18];
    const float* cw1_2 = (const float*)d_in[19];
    const float* cb1_2 = (const float*)d_in[20];
    const float* g1_2  = (const float*)d_in[21];
    const float* b1_2  = (const float*)d_in[22];
    const float* cw2_2 = (const float*)d_in[23];
    const float* cb2_2 = (const float*)d_in[24];
    const float* g2_2  = (const float*)d_in[25];
    const float* b2_2  = (const float*)d_in[26];
    const float* dsw_2 = (const float*)d_in[27];
    const float* dsb_2 = (const float*)d_in[28];

    float* T0    = (float*)d_ws;           // 34.4 MB each, rotated across blocks
    float* T1    = T0 + CH64NP;
    float* T2    = T1 + CH64NP;
    float* scale = T2 + CH64NP;
    float* shift = scale + 64;

    const dim3 cg4(NP / 16, 4);   // Cout = 64 : 4 channel tiles
    const dim3 cg1(NP / 16, 1);   // Cout = 1
    const dim3 ew(NP / 256, 64);
    const dim3 ew1(NP / 256, 1);

    // ---- Block 0: Cin=128, dil=1, downsample conv. x layout NCHW: cstride=2100, nstride=128*2100
    conv_wmma_kernel<128, 2><<<cg4, 32, 0, stream>>>(x, cw1_0, cb1_0, T0, 64, 1, 2100L, 268800L);
    bn_stats_kernel<<<64, 256, 0, stream>>>(T0, g1_0, b1_0, scale, shift);
    bn_lif_kernel<<<ew, 256, 0, stream>>>(T0, scale, shift);                    // spks1 in T0
    conv_wmma_kernel<64, 2><<<cg4, 32, 0, stream>>>(T0, cw2_0, cb2_0, T1, 64, 1, (long)NP, 2100L);
    bn_stats_kernel<<<64, 256, 0, stream>>>(T1, g2_0, b2_0, scale, shift);
    bn_lif_kernel<<<ew, 256, 0, stream>>>(T1, scale, shift);                    // spks2 in T1
    conv_wmma_kernel<128, 1><<<cg4, 32, 0, stream>>>(x, dsw_0, dsb_0, T2, 64, 0, 2100L, 268800L);
    add_lif_kernel<<<ew, 256, 0, stream>>>(T1, T2, T0);                         // block0 out -> T0

    // ---- Block 1: Cin=64, dil=2, identity residual
    conv_wmma_kernel<64, 2><<<cg4, 32, 0, stream>>>(T0, cw1_1, cb1_1, T1, 64, 2, (long)NP, 2100L);
    bn_stats_kernel<<<64, 256, 0, stream>>>(T1, g1_1, b1_1, scale, shift);
    bn_lif_kernel<<<ew, 256, 0, stream>>>(T1, scale, shift);
    conv_wmma_kernel<64, 2><<<cg4, 32, 0, stream>>>(T1, cw2_1, cb2_1, T2, 64, 2, (long)NP, 2100L);
    bn_stats_kernel<<<64, 256, 0, stream>>>(T2, g2_1, b2_1, scale, shift);
    bn_lif_kernel<<<ew, 256, 0, stream>>>(T2, scale, shift);
    add_lif_kernel<<<ew, 256, 0, stream>>>(T2, T0, T1);                         // block1 out -> T1

    // ---- Block 2: Cin=64 -> Cout=1, dil=4, downsample conv
    conv_wmma_kernel<64, 2><<<cg1, 32, 0, stream>>>(T1, cw1_2, cb1_2, T0, 1, 4, (long)NP, 2100L);
    bn_stats_kernel<<<1, 256, 0, stream>>>(T0, g1_2, b1_2, scale, shift);
    bn_lif_kernel<<<ew1, 256, 0, stream>>>(T0, scale, shift);                   // spks1 (1ch) in T0
    conv_wmma_kernel<1, 2><<<cg1, 32, 0, stream>>>(T0, cw2_2, cb2_2, T2, 1, 4, (long)NP, 2100L);
    bn_stats_kernel<<<1, 256, 0, stream>>>(T2, g2_2, b2_2, scale, shift);
    bn_lif_kernel<<<ew1, 256, 0, stream>>>(T2, scale, shift);                   // spks2 (1ch) in T2
    conv_wmma_kernel<64, 1><<<cg1, 32, 0, stream>>>(T1, dsw_2, dsb_2, T0, 1, 0, (long)NP, 2100L);
    add_lif_kernel<<<ew1, 256, 0, stream>>>(T2, T0, (float*)d_out);             // spks -> d_out[0:134400]
    tail_kernel<<<6, 256, 0, stream>>>((float*)d_out);                          // spks[:,:,-1]
}